// BrainGNN_50371376447812
// MI455X (gfx1250) — compile-verified
//
#include <hip/hip_runtime.h>
#include <hip/hip_bf16.h>

typedef __attribute__((ext_vector_type(2))) float v2f;
typedef __attribute__((ext_vector_type(8))) float v8f;

#define FDIM 64   // feature / hidden width (F == H == 64)

// ---------------- degree / norm precompute (done once, reused 3 layers) ----

__global__ void k_init_deg(float* deg, int N) {
    int i = blockIdx.x * blockDim.x + threadIdx.x;
    if (i < N) deg[i] = 1.0f;                 // self-loop contributes 1
}

__global__ void k_deg_accum(const long long* __restrict__ ei, float* deg, int E) {
    int e = blockIdx.x * blockDim.x + threadIdx.x;
    if (e < E) atomicAdd(&deg[(int)ei[E + e]], 1.0f);   // dst row of edge_index
}

__global__ void k_rsqrt(float* deg, int N) {
    int i = blockIdx.x * blockDim.x + threadIdx.x;
    if (i < N) deg[i] = rsqrtf(deg[i]);       // deg >= 1, safe
}

__global__ void k_norm(const long long* __restrict__ ei, const float* __restrict__ dinv,
                       float* __restrict__ nrm, int E) {
    int e = blockIdx.x * blockDim.x + threadIdx.x;
    if (e < E) nrm[e] = dinv[(int)ei[e]] * dinv[(int)ei[E + e]];
}

// ---------------- dense transform: C[N,64] = (relu?)A[N,64] @ W[64,64] ------
// One wave computes one 16x16 C tile with V_WMMA_F32_16X16X4_F32 (full fp32,
// no precision loss), K looped 64/4 = 16 steps, accumulator resident in VGPRs.
// Block = 128 threads = 4 waves covering the 4 column tiles of the 64-wide W.
// W (16 KB) is staged in LDS once per block; B fragments come from ds_load.
// RELU is a compile-time flag so the A-load path has no cndmask selects.

template <int RELU>
__global__ void k_gemm_wmma(const float* __restrict__ A, const float* __restrict__ W,
                            float* __restrict__ C, int N) {
    __shared__ float sW[FDIM * FDIM];

    // cooperative load of W into LDS: 128 threads x 32 iterations, b32 each
#pragma unroll
    for (int t = 0; t < (FDIM * FDIM) / 128; ++t)
        sW[t * 128 + threadIdx.x] = W[t * 128 + threadIdx.x];
    __syncthreads();

    const int lane = threadIdx.x & 31;
    const int wave = threadIdx.x >> 5;        // 0..3 -> column tile
    const int row0 = blockIdx.x * 16;
    const int col0 = wave * 16;

    // A fragment addressing (32-bit A 16x4): lanes 0-15 -> K={0,1}, lanes 16-31 -> K={2,3}
    int m = row0 + (lane & 15);
    if (m >= N) m = N - 1;                    // clamp loads; stores are guarded
    const int kb = (lane >> 4) * 2;
    const int n  = col0 + (lane & 15);
    const float* Arow = A + (long long)m * FDIM;

    v8f acc = {0.f, 0.f, 0.f, 0.f, 0.f, 0.f, 0.f, 0.f};

#pragma unroll
    for (int k0 = 0; k0 < FDIM; k0 += 4) {
        float a0 = Arow[k0 + kb];
        float a1 = Arow[k0 + kb + 1];
        if (RELU) { a0 = fmaxf(a0, 0.f); a1 = fmaxf(a1, 0.f); }
        v2f a; a.x = a0; a.y = a1;
        v2f b;
        b.x = sW[(k0 + kb) * FDIM + n];
        b.y = sW[(k0 + kb + 1) * FDIM + n];
        // (neg_a, A, neg_b, B, c_mod, C, reuse_a, reuse_b)
        acc = __builtin_amdgcn_wmma_f32_16x16x4_f32(
                false, a, false, b, (short)0, acc, false, false);
    }

    // C/D layout: VGPR r -> M = r (+8 for lanes 16-31), N = lane&15
    const int rbase = (lane >> 4) * 8;
#pragma unroll
    for (int r = 0; r < 8; ++r) {
        int row = row0 + rbase + r;
        if (row < N) C[(long long)row * FDIM + col0 + (lane & 15)] = acc[r];
    }
}

// ---------------- aggregation ----------------------------------------------
// Initialize out with the self-loop term + bias, then pure atomic scatter.

__global__ void k_self_bias(const float* __restrict__ hw, const float* __restrict__ dinv,
                            const float* __restrict__ b, float* __restrict__ out, int N) {
    int idx = blockIdx.x * blockDim.x + threadIdx.x;
    if (idx < N * FDIM) {
        int i = idx >> 6, f = idx & 63;
        float di = dinv[i];
        out[idx] = hw[idx] * di * di + b[f];
    }
}

// one thread per (edge, feature): wave32 reads 32 consecutive floats of the
// source row (coalesced, L2-resident) and scatters with f32 atomics.
__global__ void k_edge_scatter(const long long* __restrict__ ei, const float* __restrict__ nrm,
                               const float* __restrict__ hw, float* __restrict__ out, int E) {
    long long idx = (long long)blockIdx.x * blockDim.x + threadIdx.x;
    if (idx < (long long)E * FDIM) {
        int e = (int)(idx >> 6), f = (int)(idx & 63);
        int s = (int)ei[e];
        int d = (int)ei[E + e];
        atomicAdd(&out[(long long)d * FDIM + f], hw[(long long)s * FDIM + f] * nrm[e]);
    }
}

// ---------------- global mean pool -----------------------------------------

__global__ void k_pool_zero(float* pool, float* cnt, int G) {
    int idx = blockIdx.x * blockDim.x + threadIdx.x;
    if (idx < G * FDIM) pool[idx] = 0.f;
    if (idx < G) cnt[idx] = 0.f;
}

__global__ void k_pool_accum(const float* __restrict__ h, const long long* __restrict__ batch,
                             float* __restrict__ pool, float* __restrict__ cnt, int N) {
    int idx = blockIdx.x * blockDim.x + threadIdx.x;
    if (idx < N * FDIM) {
        int i = idx >> 6, f = idx & 63;
        int g = (int)batch[i];
        atomicAdd(&pool[g * FDIM + f], h[idx]);
        if (f == 0) atomicAdd(&cnt[g], 1.0f);
    }
}

__global__ void k_pool_final(const float* __restrict__ pool, const float* __restrict__ cnt,
                             float* __restrict__ out, int G) {
    int idx = blockIdx.x * blockDim.x + threadIdx.x;
    if (idx < G * FDIM) {
        int g = idx >> 6;
        out[idx] = pool[idx] / fmaxf(cnt[g], 1.0f);
    }
}

// ---------------- driver ----------------------------------------------------

extern "C" void kernel_launch(void* const* d_in, const int* in_sizes, int n_in,
                              void* d_out, int out_size, void* d_ws, size_t ws_size,
                              hipStream_t stream) {
    const float*     x     = (const float*)d_in[0];
    const long long* ei    = (const long long*)d_in[1];   // int64 edge_index [2,E]
    const long long* batch = (const long long*)d_in[2];   // int64 [N]
    const float* W1 = (const float*)d_in[3]; const float* b1 = (const float*)d_in[4];
    const float* W2 = (const float*)d_in[5]; const float* b2 = (const float*)d_in[6];
    const float* W3 = (const float*)d_in[7]; const float* b3 = (const float*)d_in[8];

    const int N = in_sizes[0] / FDIM;
    const int E = in_sizes[1] / 2;
    const int G = out_size / FDIM;
    float* out = (float*)d_out;

    // workspace layout (all L2-resident working set, ~42 MB of 192 MB L2)
    float* ws   = (float*)d_ws;
    float* dinv = ws;                 ws += N;
    float* nrm  = ws;                 ws += E;
    float* bufA = ws;                 ws += (size_t)N * FDIM;  // GEMM output (hw)
    float* bufB = ws;                 ws += (size_t)N * FDIM;  // layer outputs (ping)
    float* bufC = ws;                 ws += (size_t)N * FDIM;  // layer outputs (pong)
    float* pool = ws;                 ws += (size_t)G * FDIM;
    float* cnt  = ws;                 ws += G;

    const int T = 256;
    const int gN  = (N + T - 1) / T;
    const int gE  = (E + T - 1) / T;
    const int gNF = (N * FDIM + T - 1) / T;
    const long long totEF = (long long)E * FDIM;
    const int gEF = (int)((totEF + T - 1) / T);
    const int gG  = (G * FDIM + T - 1) / T;
    const int gMM = (N + 15) / 16;    // 16-row WMMA tiles

    // normalization (computed once, reused for all 3 layers)
    k_init_deg  <<<gN, T, 0, stream>>>(dinv, N);
    k_deg_accum <<<gE, T, 0, stream>>>(ei, dinv, E);
    k_rsqrt    <<<gN, T, 0, stream>>>(dinv, N);
    k_norm      <<<gE, T, 0, stream>>>(ei, dinv, nrm, E);

    // layer 1: x -> bufB
    k_gemm_wmma<0><<<gMM, 128, 0, stream>>>(x,    W1, bufA, N);
    k_self_bias   <<<gNF, T,   0, stream>>>(bufA, dinv, b1, bufB, N);
    k_edge_scatter<<<gEF, T,   0, stream>>>(ei, nrm, bufA, bufB, E);

    // layer 2: relu(bufB) -> bufC   (ReLU fused into GEMM A-load, compile-time)
    k_gemm_wmma<1><<<gMM, 128, 0, stream>>>(bufB, W2, bufA, N);
    k_self_bias   <<<gNF, T,   0, stream>>>(bufA, dinv, b2, bufC, N);
    k_edge_scatter<<<gEF, T,   0, stream>>>(ei, nrm, bufA, bufC, E);

    // layer 3: relu(bufC) -> bufB   (no ReLU on output)
    k_gemm_wmma<1><<<gMM, 128, 0, stream>>>(bufC, W3, bufA, N);
    k_self_bias   <<<gNF, T,   0, stream>>>(bufA, dinv, b3, bufB, N);
    k_edge_scatter<<<gEF, T,   0, stream>>>(ei, nrm, bufA, bufB, E);

    // global mean pool
    k_pool_zero  <<<gG,  T, 0, stream>>>(pool, cnt, G);
    k_pool_accum <<<gNF, T, 0, stream>>>(bufB, batch, pool, cnt, N);
    k_pool_final <<<gG,  T, 0, stream>>>(pool, cnt, out, G);
}